// ExplicitMem_61950608278068
// MI455X (gfx1250) — compile-verified
//
#include <hip/hip_runtime.h>
#include <math.h>

#define BS 256
#define R 4
#define W 128
#define N 2048
#define H 256
#define NT 256
#define NWAVE 8
#define NC 128
#define NCH (N / NC)

static_assert(NT == 256 && NC == 128, "layout assumptions");

constexpr float TAU_INV = 1.0f;   // TAU = 1.0
constexpr float ALPHA   = 0.7f;
constexpr float EPS     = 1e-8f;

typedef __attribute__((ext_vector_type(2))) float v2f;
typedef __attribute__((ext_vector_type(8))) float v8f;

__device__ __forceinline__ float gumbelf(float u) {
    return -__logf(-__logf(u + 1e-10f) + 1e-10f);
}
__device__ __forceinline__ float wave_sum(float v) {
    #pragma unroll
    for (int m = 16; m > 0; m >>= 1) v += __shfl_xor(v, m, 32);
    return v;
}
__device__ __forceinline__ float wave_max(float v) {
    #pragma unroll
    for (int m = 16; m > 0; m >>= 1) v = fmaxf(v, __shfl_xor(v, m, 32));
    return v;
}

__global__ __launch_bounds__(NT) void dnc_step_kernel(
    const float* __restrict__ k_r,    const float* __restrict__ m_t,
    const float* __restrict__ hx,     const float* __restrict__ memory,
    const float* __restrict__ usage,  const float* __restrict__ read_w,
    const float* __restrict__ Wg,     const float* __restrict__ bg,
    const float* __restrict__ noise_w,const float* __restrict__ noise_r,
    float* __restrict__ o_mread, float* __restrict__ o_mem,
    float* __restrict__ o_usage, float* __restrict__ o_rw,
    float* __restrict__ o_mer)
{
    // LDS: ~123 KB total
    __shared__ float s_mt[W * (NC + 1)];     // staged mem_new chunk, [w][n_local], padded
    __shared__ float s_lr[R * (N + 1)];      // read logits -> read_wt, padded rows
    __shared__ float s_lw[N];                // write logits -> write_wt
    __shared__ float s_ut[N];                // u_t
    __shared__ float s_rkp[16 * (W + 1)];    // normalized read keys, rows 4..15 zero, padded
    __shared__ float s_wm[W];                // tanh(m_t)  (un-normalized, used for update)
    __shared__ float s_wmn[W];               // normalized write key
    __shared__ float s_s2[NT];               // per-thread column sumsq partials
    __shared__ float s_invn[NC];             // 1/(||col||+eps) for current chunk
    __shared__ float s_red[NWAVE];           // block reduction scratch
    __shared__ float s_scal[2];              // [0]=gamma  [1]=reduction result

    const int b    = blockIdx.x;
    const int tid  = threadIdx.x;
    const int lane = tid & 31;
    const int wave = tid >> 5;
    const size_t bN = (size_t)b * N;

    // ---- Step 1: keys (tanh + norms), gamma: per-wave roles ----
    if (wave < R) {
        float ss = 0.f;
        for (int w = lane; w < W; w += 32) {
            float v = tanhf(k_r[(size_t)b * (R * W) + wave * W + w]);
            s_rkp[wave * (W + 1) + w] = v;
            ss += v * v;
        }
        ss = wave_sum(ss);
        if (lane == 0) s_red[wave] = sqrtf(ss) + EPS;
    } else if (wave == 4) {
        float ss = 0.f;
        for (int w = lane; w < W; w += 32) {
            float v = tanhf(m_t[(size_t)b * W + w]);
            s_wm[w] = v;
            ss += v * v;
        }
        ss = wave_sum(ss);
        if (lane == 0) s_red[4] = sqrtf(ss) + EPS;
    } else if (wave == 5) {
        float dot = 0.f;
        for (int h = lane; h < H; h += 32)
            dot += hx[(size_t)b * H + h] * Wg[h];
        dot = wave_sum(dot);
        if (lane == 0) s_scal[0] = 1.f / (1.f + __expf(-(dot + bg[0])));
    }
    __syncthreads();

    // normalize write key / read keys, zero pad rows 4..15 of s_rkp
    for (int i = tid; i < W; i += NT) s_wmn[i] = s_wm[i] / s_red[4];
    for (int i = tid; i < R * W; i += NT) {
        int r = i >> 7, w = i & (W - 1);
        s_rkp[r * (W + 1) + w] /= s_red[r];
    }
    for (int i = tid; i < 12 * (W + 1); i += NT)
        s_rkp[4 * (W + 1) + i] = 0.f;

    // ---- Step 2: u_t = alpha*usage + (1-alpha)*sum_r read_weights ----
    for (int n = tid; n < N; n += NT) {
        const size_t base = (size_t)b * R * N + n;
        float rs = read_w[base] + read_w[base + N] + read_w[base + 2 * N] + read_w[base + 3 * N];
        s_ut[n] = ALPHA * usage[bN + n] + (1.f - ALPHA) * rs;
    }
    __syncthreads();
    const float gamma = s_scal[0];

    // ---- Step 3: write logits: (1 + dist - gamma*u_t + gumbel)/TAU ----
    const float* memb = memory + (size_t)b * W * N;
    for (int n = tid; n < N; n += NT) {
        float s1 = 0.f, s2 = 0.f;
        for (int w = 0; w < W; ++w) {
            float v = memb[(size_t)w * N + n];
            s1 = fmaf(s_wmn[w], v, s1);
            s2 = fmaf(v, v, s2);
        }
        float dist = s1 / (sqrtf(s2) + EPS);
        float g = gumbelf(noise_w[bN + n]);
        s_lw[n] = (1.f + dist - gamma * s_ut[n] + g) * TAU_INV;
    }
    __syncthreads();

    // ---- Step 4: write softmax -> write_wt; usage_new ----
    {
        float mx = -INFINITY;
        for (int n = tid; n < N; n += NT) mx = fmaxf(mx, s_lw[n]);
        mx = wave_max(mx);
        if (lane == 0) s_red[wave] = mx;
        __syncthreads();
        if (tid == 0) {
            float m = s_red[0];
            for (int i = 1; i < NWAVE; ++i) m = fmaxf(m, s_red[i]);
            s_scal[1] = m;
        }
        __syncthreads();
        mx = s_scal[1];
        float zs = 0.f;
        for (int n = tid; n < N; n += NT) {
            float e = __expf(s_lw[n] - mx);
            s_lw[n] = e;
            zs += e;
        }
        zs = wave_sum(zs);
        if (lane == 0) s_red[wave] = zs;
        __syncthreads();
        if (tid == 0) {
            float z = 0.f;
            for (int i = 0; i < NWAVE; ++i) z += s_red[i];
            s_scal[1] = 1.f / z;
        }
        __syncthreads();
        const float iz = s_scal[1];
        for (int n = tid; n < N; n += NT) {
            float wwt = s_lw[n] * iz;
            s_lw[n] = wwt;
            o_usage[bN + n] = s_ut[n] * (1.f - wwt);
        }
        __syncthreads();
    }

    // ---- Step 5: pass B — mem_new + read logits via fp32 WMMA ----
    float* omem_b = o_mem + (size_t)b * W * N;
    for (int ch = 0; ch < NCH; ++ch) {
        const int nb = ch * NC;
        {   // 5a: generate chunk; 2 threads per column (split w)
            const int c  = tid & (NC - 1);
            const int hh = tid >> 7;            // 0 or 1
            const int n  = nb + c;
            const float wwt = s_lw[n];
            const float omw = 1.f - wwt;
            float s2 = 0.f;
            for (int w = hh * 64; w < hh * 64 + 64; ++w) {
                float v  = memb[(size_t)w * N + n];
                float mn = fmaf(v, omw, s_wm[w] * wwt);
                omem_b[(size_t)w * N + n] = mn;
                s_mt[w * (NC + 1) + c] = mn;
                s2 = fmaf(mn, mn, s2);
            }
            s_s2[tid] = s2;
        }
        __syncthreads();
        if (tid < NC)
            s_invn[tid] = 1.f / (sqrtf(s_s2[tid] + s_s2[NC + tid]) + EPS);
        __syncthreads();
        {   // 5b: one 16-col tile per wave: D = rk_norm(16x128) x mem_new(128x16)
            const int col  = lane & 15;
            const int koff = (lane >> 4) * 2;
            const int nl   = wave * 16 + col;
            v8f acc = {};
            #pragma unroll
            for (int kk = 0; kk < W / 4; ++kk) {
                const int k0 = 4 * kk + koff;
                v2f av, bv;
                av.x = s_rkp[(lane & 15) * (W + 1) + k0];
                av.y = s_rkp[(lane & 15) * (W + 1) + k0 + 1];
                bv.x = s_mt[(k0)     * (NC + 1) + nl];
                bv.y = s_mt[(k0 + 1) * (NC + 1) + nl];
                acc = __builtin_amdgcn_wmma_f32_16x16x4_f32(
                    false, av, false, bv, (short)0, acc, false, false);
            }
            if (lane < 16) {
                const int n = nb + nl;
                const float invn = s_invn[nl];
                #pragma unroll
                for (int r = 0; r < R; ++r) {
                    float g = gumbelf(noise_r[(size_t)b * R * N + (size_t)r * N + n]);
                    s_lr[r * (N + 1) + n] = (acc[r] * invn + g) * TAU_INV;
                }
            }
        }
        __syncthreads();
    }

    // ---- Step 6: m_erased[w] = sum_n write_wt[n] * memory[w,n]  (L2-hot rows) ----
    for (int w = wave; w < W; w += NWAVE) {
        const float* mrow = memb + (size_t)w * N;
        float acc = 0.f;
        for (int n = lane; n < N; n += 32)
            acc = fmaf(s_lw[n], mrow[n], acc);
        acc = wave_sum(acc);
        if (lane == 0) o_mer[(size_t)b * W + w] = acc;
    }

    // ---- Step 7: four read softmaxes -> read_wt ----
    for (int r = 0; r < R; ++r) {
        float* lrow = s_lr + r * (N + 1);
        float mx = -INFINITY;
        for (int n = tid; n < N; n += NT) mx = fmaxf(mx, lrow[n]);
        mx = wave_max(mx);
        if (lane == 0) s_red[wave] = mx;
        __syncthreads();
        if (tid == 0) {
            float m = s_red[0];
            for (int i = 1; i < NWAVE; ++i) m = fmaxf(m, s_red[i]);
            s_scal[1] = m;
        }
        __syncthreads();
        mx = s_scal[1];
        float zs = 0.f;
        for (int n = tid; n < N; n += NT) {
            float e = __expf(lrow[n] - mx);
            lrow[n] = e;
            zs += e;
        }
        zs = wave_sum(zs);
        if (lane == 0) s_red[wave] = zs;
        __syncthreads();
        if (tid == 0) {
            float z = 0.f;
            for (int i = 0; i < NWAVE; ++i) z += s_red[i];
            s_scal[1] = 1.f / z;
        }
        __syncthreads();
        const float iz = s_scal[1];
        for (int n = tid; n < N; n += NT) {
            float rw = lrow[n] * iz;
            lrow[n] = rw;
            o_rw[(size_t)b * R * N + (size_t)r * N + n] = rw;
        }
        __syncthreads();
    }

    // ---- Step 8: pass C — m_read via fp32 WMMA, K over n ----
    v8f accm = {};
    const int wt = wave * 16;                 // this wave's 16 w-columns
    for (int ch = 0; ch < NCH; ++ch) {
        const int nb = ch * NC;
        {   // 8a: stage mem_new chunk (L2-hot)
            const int c  = tid & (NC - 1);
            const int hh = tid >> 7;
            const int n  = nb + c;
            for (int w = hh * 64; w < hh * 64 + 64; ++w)
                s_mt[w * (NC + 1) + c] = omem_b[(size_t)w * N + n];
        }
        __syncthreads();
        {   // 8b: D(16 r x 16 w) += read_wt(16 x NC) x mem_new^T(NC x 16)
            const int row  = lane & 15;
            const int koff = (lane >> 4) * 2;
            const int arow = (row < R) ? row : 0;
            const bool rv  = (row < R);
            const int col  = lane & 15;
            #pragma unroll
            for (int kk = 0; kk < NC / 4; ++kk) {
                const int k0 = 4 * kk + koff;
                v2f av, bv;
                float a0 = s_lr[arow * (N + 1) + nb + k0];
                float a1 = s_lr[arow * (N + 1) + nb + k0 + 1];
                av.x = rv ? a0 : 0.f;
                av.y = rv ? a1 : 0.f;
                bv.x = s_mt[(wt + col) * (NC + 1) + k0];
                bv.y = s_mt[(wt + col) * (NC + 1) + k0 + 1];
                accm = __builtin_amdgcn_wmma_f32_16x16x4_f32(
                    false, av, false, bv, (short)0, accm, false, false);
            }
        }
        __syncthreads();
    }
    if (lane < 16) {
        #pragma unroll
        for (int r = 0; r < R; ++r)
            o_mread[(size_t)b * R * W + (size_t)r * W + wt + lane] = accm[r];
    }
}

extern "C" void kernel_launch(void* const* d_in, const int* in_sizes, int n_in,
                              void* d_out, int out_size, void* d_ws, size_t ws_size,
                              hipStream_t stream) {
    (void)in_sizes; (void)n_in; (void)out_size; (void)d_ws; (void)ws_size;
    const float* k_r     = (const float*)d_in[0];
    const float* m_t     = (const float*)d_in[1];
    const float* hx      = (const float*)d_in[2];
    const float* memory  = (const float*)d_in[3];
    const float* usage   = (const float*)d_in[4];
    const float* read_w  = (const float*)d_in[5];
    const float* Wg      = (const float*)d_in[6];
    const float* bg      = (const float*)d_in[7];
    const float* noise_w = (const float*)d_in[8];
    const float* noise_r = (const float*)d_in[9];
    // d_in[10] = m_idx (int64), d_in[11] = seq_idx : unused by the reference math

    float* out      = (float*)d_out;
    float* o_mread  = out;                                   // (BS,R,W)
    float* o_mem    = o_mread + (size_t)BS * R * W;          // (BS,W,N)
    float* o_usage  = o_mem   + (size_t)BS * W * N;          // (BS,N)
    float* o_rw     = o_usage + (size_t)BS * N;              // (BS,R,N)
    float* o_mer    = o_rw    + (size_t)BS * R * N;          // (BS,1,W)

    dnc_step_kernel<<<BS, NT, 0, stream>>>(k_r, m_t, hx, memory, usage, read_w,
                                           Wg, bg, noise_w, noise_r,
                                           o_mread, o_mem, o_usage, o_rw, o_mer);
}